// LSTMClassification_55525337202792
// MI455X (gfx1250) — compile-verified
//
#include <hip/hip_runtime.h>

typedef __attribute__((ext_vector_type(2))) float v2f;
typedef __attribute__((ext_vector_type(8))) float v8f;

#define T_LEN 1024

__device__ __forceinline__ float tanh_fast(float v) {
#if __has_builtin(__builtin_amdgcn_tanhf)
    return __builtin_amdgcn_tanhf(v);   // v_tanh_f32 (gfx1250 TRANS op)
#else
    return tanhf(v);
#endif
}

// One wave processes 16 sequences (columns). Per timestep:
//   D[m, n] = sum_k A[m,k] * B[k,n] + C[m,n]   via v_wmma_f32_16x16x4_f32
//
// Gate rows are PERMUTED so each lane half owns all four gates of two
// hidden units:
//   rows 0-7  (lanes 0-15):  i0,i1,f0,f1,g0,g1,o0,o1   -> state (c0,c1,h0,h1)
//   rows 8-15 (lanes 16-31): i2,i3,f2,f3,g2,g3,o2,o3   -> state (c2,c3,h2,h3)
// This makes the LSTM cell update half-local (no cross-lane traffic) and the
// B matrix (4x16: VGPR0 = K=0 / K=2 per half, VGPR1 = K=1 / K=3) is exactly
// {h_a, h_b} per half with no selects.
//
// Sigmoid gates use sigma(x) = 0.5*tanh(0.5x)+0.5 with the 0.5 folded into
// the corresponding rows of W_hh / W_ih / biases.
__global__ __launch_bounds__(256) void lstm_wmma_f32_kernel(
    const float* __restrict__ x,
    const float* __restrict__ W_ih,
    const float* __restrict__ W_hh,
    const float* __restrict__ b_ih,
    const float* __restrict__ b_hh,
    const float* __restrict__ fc_w,
    const float* __restrict__ fc_b,
    float* __restrict__ out,
    int B)
{
    const int lane = threadIdx.x & 31;
    const int wave = threadIdx.x >> 5;
    const bool lo  = lane < 16;          // lower half-wave?
    const int  n   = lane & 15;          // batch column within wave
    const int  col = (blockIdx.x * 8 + wave) * 16 + n;   // sequence index

    // permuted-row -> original gate-row map (i=0-3, f=4-7, g=8-11, o=12-15)
    const int P[16] = {0,1,4,5,8,9,12,13,  2,3,6,7,10,11,14,15};

    // ---- A matrix (16x4, M = permuted gate row, K = hidden unit):
    //      lanes 0-15 hold K=0,1; lanes 16-31 hold K=2,3 (ISA 7.12.2).
    const int   ag   = P[n];
    const float ascl = (ag >= 8 && ag < 12) ? 1.0f : 0.5f;  // tanh rows unscaled
    const int   k0   = lo ? 0 : 2;
    v2f A;
    A.x = ascl * W_hh[ag * 4 + k0];
    A.y = ascl * W_hh[ag * 4 + k0 + 1];

    // ---- per-lane constants for the C (input-gate) matrix rows
    float wih_s[8], bias_s[8];
#pragma unroll
    for (int v = 0; v < 8; ++v) {
        const int   gr = P[v + (lo ? 0 : 8)];
        const float s  = (gr >= 8 && gr < 12) ? 1.0f : 0.5f;
        wih_s[v]  = s * W_ih[gr];
        bias_s[v] = s * (b_ih[gr] + b_hh[gr]);
    }

    // state: this half's two hidden units (a = unit 0 or 2, b = unit 1 or 3)
    float ha = 0.f, hb = 0.f;
    float ca = 0.f, cb = 0.f;

    const float* xp = x + (size_t)col * T_LEN;

    for (int t = 0; t < T_LEN; t += 4) {
        const float4 xq = *(const float4*)(xp + t);
        const float xs[4] = {xq.x, xq.y, xq.z, xq.w};
#pragma unroll
        for (int j = 0; j < 4; ++j) {
            const float xv = xs[j];

            // C = x*W_ih + bias (pre-scaled), 8 permuted gate rows per lane
            v8f C;
#pragma unroll
            for (int v = 0; v < 8; ++v) C[v] = fmaf(xv, wih_s[v], bias_s[v]);

            // B (4x16): direct from half-local state, no selects
            v2f Bm;
            Bm.x = ha;
            Bm.y = hb;

            const v8f D = __builtin_amdgcn_wmma_f32_16x16x4_f32(
                false, A, false, Bm, (short)0, C, false, false);

            // activations (per-VGPR compile-time affine; v4,v5 are raw tanh)
            const float i_a = fmaf(tanh_fast(D[0]), 0.5f, 0.5f);
            const float i_b = fmaf(tanh_fast(D[1]), 0.5f, 0.5f);
            const float f_a = fmaf(tanh_fast(D[2]), 0.5f, 0.5f);
            const float f_b = fmaf(tanh_fast(D[3]), 0.5f, 0.5f);
            const float g_a = tanh_fast(D[4]);
            const float g_b = tanh_fast(D[5]);
            const float o_a = fmaf(tanh_fast(D[6]), 0.5f, 0.5f);
            const float o_b = fmaf(tanh_fast(D[7]), 0.5f, 0.5f);

            // half-local cell update
            ca = fmaf(f_a, ca, i_a * g_a);
            cb = fmaf(f_b, cb, i_b * g_b);
            ha = o_a * tanh_fast(ca);
            hb = o_b * tanh_fast(cb);
        }
    }

    // final FC: out[col] = h . fc_w + fc_b
    // lo half has (h0,h1), hi half has (h2,h3): one cross-half reduce at the end
    const float wA = lo ? fc_w[0] : fc_w[2];
    const float wB = lo ? fc_w[1] : fc_w[3];
    const float part = fmaf(ha, wA, hb * wB);
    const float other = __shfl_xor(part, 16, 32);
    if (lo && col < B) {
        out[col] = part + other + fc_b[0];
    }
}

extern "C" void kernel_launch(void* const* d_in, const int* in_sizes, int n_in,
                              void* d_out, int out_size, void* d_ws, size_t ws_size,
                              hipStream_t stream) {
    const float* x    = (const float*)d_in[0];
    const float* W_ih = (const float*)d_in[1];
    const float* W_hh = (const float*)d_in[2];
    const float* b_ih = (const float*)d_in[3];
    const float* b_hh = (const float*)d_in[4];
    const float* fc_w = (const float*)d_in[5];
    const float* fc_b = (const float*)d_in[6];
    float* out = (float*)d_out;

    const int B = in_sizes[0] / T_LEN;      // I == 1
    const int seq_per_block = 8 * 16;       // 8 waves * 16 sequences
    const int grid = (B + seq_per_block - 1) / seq_per_block;

    lstm_wmma_f32_kernel<<<grid, 256, 0, stream>>>(
        x, W_ih, W_hh, b_ih, b_hh, fc_w, fc_b, out, B);
}